// VQCodebook_5574867550331
// MI455X (gfx1250) — compile-verified
//
#include <hip/hip_runtime.h>
#include <hip/hip_bf16.h>
#include <math.h>

typedef float v2f __attribute__((ext_vector_type(2)));
typedef float v8f __attribute__((ext_vector_type(8)));

// ---- constants -------------------------------------------------------------
#define BATCH 256
#define DIM 128
#define KCODES 8192
// d_out layout: recon[256*3*64*64], codebook_loss, commitment_loss, idx[256]
#define RECON_N   (256*3*64*64)       // 3145728
#define LOSS0_OFF RECON_N
#define LOSS1_OFF (RECON_N + 1)
#define IDX_OFF   (RECON_N + 2)

// ---- f32 WMMA 16x16x4 helper ----------------------------------------------
__device__ inline v8f wmma_f32_4(v2f a, v2f b, v8f c) {
  // (neg_a, A, neg_b, B, c_mod, C, reuse_a, reuse_b)
  return __builtin_amdgcn_wmma_f32_16x16x4_f32(false, a, false, b, (short)0, c,
                                               false, false);
}

// ============================================================================
// Kernel A: fused conv1(3->64,SAME)+ReLU and 9-region sums S[b, i*9+ky*3+kx].
// One block per batch image; image staged via async global->LDS DMA; each
// thread processes 4-pixel quads so one LDS weight read feeds 4 FMAs.
// ============================================================================
__global__ void __launch_bounds__(256)
kconv1_sums(const float* __restrict__ patch, const float* __restrict__ w1,
            const float* __restrict__ b1, float* __restrict__ Sbuf) {
  __shared__ __align__(16) float x[3 * 64 * 64];   // input image
  __shared__ float w[64 * 27];
  __shared__ float bb[64];
  __shared__ float aT[64], aR0[64], aR63[64], aC0[64], aC63[64];
  __shared__ float corn[4][64];      // (0,0) (0,63) (63,0) (63,63)
  const int b = blockIdx.x, t = threadIdx.x;

  // ---- async stage of input image into LDS: 768 b128 transfers -------------
  {
    const unsigned lbase = (unsigned)(unsigned long long)(&x[0]);
    const float* gsrc = patch + b * 12288;
#pragma unroll
    for (int it = 0; it < 3; ++it) {
      const int i16 = t + it * 256;                 // float4 index
      const unsigned ldst = lbase + (unsigned)(i16 * 16);
      const unsigned long long ga = (unsigned long long)(gsrc + i16 * 4);
      asm volatile("global_load_async_to_lds_b128 %0, %1, off"
                   :: "v"(ldst), "v"(ga) : "memory");
    }
  }
  for (int i = t; i < 1728; i += 256) w[i] = w1[i];
  if (t < 64) {
    bb[t] = b1[t];
    aT[t] = aR0[t] = aR63[t] = aC0[t] = aC63[t] = 0.f;
    corn[0][t] = corn[1][t] = corn[2][t] = corn[3][t] = 0.f;
  }
  asm volatile("s_wait_asynccnt 0x0" ::: "memory");
  __syncthreads();

  float accT[64];
#pragma unroll
  for (int j = 0; j < 64; ++j) accT[j] = 0.f;

  // 1024 quads of 4 horizontally adjacent pixels
  for (int g = t; g < 1024; g += 256) {
    const int r = g >> 4;            // row 0..63
    const int c4 = (g & 15) * 4;     // col 0,4,...,60
    float xv[54];                    // [ci][dy][6] window (cols c4-1 .. c4+4)
#pragma unroll
    for (int ci = 0; ci < 3; ++ci)
#pragma unroll
      for (int dy = 0; dy < 3; ++dy)
#pragma unroll
        for (int u = 0; u < 6; ++u) {
          const int rr = r + dy - 1, cc = c4 - 1 + u;
          xv[ci * 18 + dy * 6 + u] =
              (rr >= 0 && rr < 64 && cc >= 0 && cc < 64)
                  ? x[ci * 4096 + rr * 64 + cc] : 0.f;
        }
    const bool r0 = (r == 0), r63 = (r == 63);
    const bool cL = (c4 == 0), cR = (c4 == 60);
#pragma unroll
    for (int j = 0; j < 64; ++j) {
      float h0 = bb[j], h1 = h0, h2 = h0, h3 = h0;
#pragma unroll
      for (int ci = 0; ci < 3; ++ci)
#pragma unroll
        for (int dy = 0; dy < 3; ++dy)
#pragma unroll
          for (int dx = 0; dx < 3; ++dx) {
            const float wv = w[j * 27 + ci * 9 + dy * 3 + dx];
            const int base = ci * 18 + dy * 6 + dx;
            h0 = fmaf(wv, xv[base + 0], h0);
            h1 = fmaf(wv, xv[base + 1], h1);
            h2 = fmaf(wv, xv[base + 2], h2);
            h3 = fmaf(wv, xv[base + 3], h3);
          }
      h0 = fmaxf(h0, 0.f); h1 = fmaxf(h1, 0.f);
      h2 = fmaxf(h2, 0.f); h3 = fmaxf(h3, 0.f);
      const float hs = (h0 + h1) + (h2 + h3);
      accT[j] += hs;
      if (r0)  atomicAdd(&aR0[j], hs);
      if (r63) atomicAdd(&aR63[j], hs);
      if (cL)  atomicAdd(&aC0[j], h0);
      if (cR)  atomicAdd(&aC63[j], h3);
      if (r0 && cL)  corn[0][j] = h0;
      if (r0 && cR)  corn[1][j] = h3;
      if (r63 && cL) corn[2][j] = h0;
      if (r63 && cR) corn[3][j] = h3;
    }
  }
#pragma unroll
  for (int j = 0; j < 64; ++j) atomicAdd(&aT[j], accT[j]);
  __syncthreads();

  if (t < 64) {
    const int j = t;
    const float T = aT[j], R0 = aR0[j], R63 = aR63[j], C0 = aC0[j], C63 = aC63[j];
    float S[9];
    S[0] = T - R63 - C63 + corn[3][j];  // ky=0,kx=0: rows0..62, cols0..62
    S[1] = T - R63;                     // ky=0,kx=1
    S[2] = T - R63 - C0 + corn[2][j];   // ky=0,kx=2
    S[3] = T - C63;                     // ky=1,kx=0
    S[4] = T;                           // ky=1,kx=1
    S[5] = T - C0;                      // ky=1,kx=2
    S[6] = T - R0 - C63 + corn[1][j];   // ky=2,kx=0: rows1..63
    S[7] = T - R0;                      // ky=2,kx=1
    S[8] = T - R0 - C0 + corn[0][j];    // ky=2,kx=2
    for (int q = 0; q < 9; ++q) Sbuf[b * 576 + j * 9 + q] = S[q];
  }
}

// ============================================================================
// Prep kernels: effective conv2 weight [576,128], codebook norms, decoder
// layer-1 category-summed weights WS1[9][64][128], decoder layer-2/3 weight
// flip/transpose, and argmin accumulator init.
// ============================================================================
__global__ void kprep_w2(const float* __restrict__ ew2, float* __restrict__ W2eff) {
  const int tid = blockIdx.x * 256 + threadIdx.x;
  if (tid >= 576 * 128) return;
  const int kidx = tid >> 7, o = tid & 127;
  const int i = kidx / 9, tt = kidx % 9;
  W2eff[kidx * 128 + o] = ew2[(o * 64 + i) * 9 + tt] * (1.0f / 4096.0f);
}

__global__ void kprep_en(const float* __restrict__ emb, float* __restrict__ en) {
  const int k = blockIdx.x * 256 + threadIdx.x;
  if (k >= KCODES) return;
  const float4* e = (const float4*)(emb + k * 128);
  float s = 0.f;
  for (int i = 0; i < 32; ++i) {
    float4 v = e[i];
    s += v.x * v.x + v.y * v.y + v.z * v.z + v.w * v.w;
  }
  en[k] = s;
}

__global__ void kprep_wt(const float* __restrict__ dw1, const float* __restrict__ dw2,
                         const float* __restrict__ dw3, float* __restrict__ WS1,
                         float* __restrict__ wt2, float* __restrict__ wt3) {
  const int tid = blockIdx.x * 256 + threadIdx.x;
  if (tid < 73728) {
    // WS1[(cat*64+o)*128+i] = sum over valid flipped taps of ConvT1 weight.
    const int cat = tid / 8192, rem = tid % 8192;
    const int o = rem >> 7, i = rem & 127;
    const int catr = cat / 3, catc = cat % 3;
    const int ky0 = (catr == 0) ? 1 : 0, ky1 = (catr == 2) ? 1 : 2;
    const int kx0 = (catc == 0) ? 1 : 0, kx1 = (catc == 2) ? 1 : 2;
    float s = 0.f;
    for (int ky = ky0; ky <= ky1; ++ky)
      for (int kx = kx0; kx <= kx1; ++kx)
        s += dw1[(i * 64 + o) * 9 + (2 - ky) * 3 + (2 - kx)];
    WS1[tid] = s;
  } else if (tid < 73728 + 36864) {        // wt2[(o*9+t)*64+i]
    const int u = tid - 73728;
    const int o = u / 576, rem = u % 576;
    const int tt = rem / 64, i = rem % 64;
    const int ky = tt / 3, kx = tt % 3;
    wt2[u] = dw2[(i * 64 + o) * 9 + (2 - ky) * 3 + (2 - kx)];
  } else if (tid < 73728 + 36864 + 1728) { // wt3[(o*9+t)*64+i], dw3 [64][3][3][3]
    const int u = tid - 73728 - 36864;
    const int o = u / 576, rem = u % 576;
    const int tt = rem / 64, i = rem % 64;
    const int ky = tt / 3, kx = tt % 3;
    wt3[u] = dw3[(i * 3 + o) * 9 + (2 - ky) * 3 + (2 - kx)];
  }
}

__global__ void kinit_best(unsigned long long* best) { best[threadIdx.x] = ~0ull; }

// ============================================================================
// Kernel B: z[256,128] = Sbuf[256,576] @ W2eff[576,128] + b2 via f32 WMMA.
// One wave per 16x16 tile; grid (16, 8).
// ============================================================================
__global__ void __launch_bounds__(32)
kz_gemm(const float* __restrict__ S, const float* __restrict__ W2eff,
        const float* __restrict__ b2, float* __restrict__ z) {
  const int tm = blockIdx.x, tn = blockIdx.y;
  const int lane = threadIdx.x & 31;
  const int m = lane & 15, hi = lane >> 4, kh = hi * 2;
  v8f acc = {};
  const float* Arow = S + (tm * 16 + m) * 576;
  for (int k0 = 0; k0 < 576; k0 += 4) {
    v2f a, bfr;
    a.x = Arow[k0 + kh];
    a.y = Arow[k0 + kh + 1];
    const int n = tn * 16 + m;  // B col index == lane&15
    bfr.x = W2eff[(k0 + kh) * 128 + n];
    bfr.y = W2eff[(k0 + kh + 1) * 128 + n];
    acc = wmma_f32_4(a, bfr, acc);
  }
  const int n = tn * 16 + (lane & 15);
  const float bias = b2[n];
#pragma unroll
  for (int v = 0; v < 8; ++v) {
    const int mm = tm * 16 + v + hi * 8;
    z[mm * 128 + n] = acc[v] + bias;
  }
}

// ============================================================================
// Kernel C: VQ nearest codebook. score = ||e_k||^2 - 2 z.e_k (drop ||z||^2).
// WMMA f32 K-loop, then packed (ordered-float<<32 | k) u64 atomicMin.
// grid (16 row-tiles, 128 chunks of 64 codes), 1 wave per block.
// ============================================================================
__global__ void __launch_bounds__(32)
kvq(const float* __restrict__ z, const float* __restrict__ emb,
    const float* __restrict__ en, unsigned long long* __restrict__ best) {
  __shared__ unsigned long long lb[16];
  const int tm = blockIdx.x;
  const int nbase = blockIdx.y * 64;
  const int lane = threadIdx.x & 31;
  const int m = lane & 15, hi = lane >> 4, kh = hi * 2;
  if (lane < 16) lb[lane] = ~0ull;
  __syncthreads();

  v8f acc[4] = {};
  const float* Arow = z + (tm * 16 + m) * 128;
  for (int k0 = 0; k0 < 128; k0 += 4) {
    v2f a;
    a.x = Arow[k0 + kh];
    a.y = Arow[k0 + kh + 1];
#pragma unroll
    for (int t4 = 0; t4 < 4; ++t4) {
      const int n = nbase + t4 * 16 + m;  // B[k][n] = emb[n][k]
      v2f bfr;
      bfr.x = emb[n * 128 + k0 + kh];
      bfr.y = emb[n * 128 + k0 + kh + 1];
      acc[t4] = wmma_f32_4(a, bfr, acc[t4]);
    }
  }
#pragma unroll
  for (int t4 = 0; t4 < 4; ++t4) {
    const int n = nbase + t4 * 16 + (lane & 15);
    const float e = en[n];
#pragma unroll
    for (int v = 0; v < 8; ++v) {
      const int mrow = v + hi * 8;
      const float score = e - 2.0f * acc[t4][v];
      const unsigned bits = __float_as_uint(score);
      const unsigned key = (bits & 0x80000000u) ? ~bits : (bits | 0x80000000u);
      const unsigned long long pk =
          ((unsigned long long)key << 32) | (unsigned long long)(unsigned)n;
      atomicMin(&lb[mrow], pk);
    }
  }
  __syncthreads();
  if (lane < 16) atomicMin(&best[tm * 16 + lane], lb[lane]);
}

// ============================================================================
// Kernel D: gather z_q, write indices, compute both losses (forward-identical).
// ============================================================================
__global__ void __launch_bounds__(256)
kfinalize(const unsigned long long* __restrict__ best, const float* __restrict__ emb,
          const float* __restrict__ z, float* __restrict__ zq,
          float* __restrict__ out) {
  __shared__ float red[256];
  const int t = threadIdx.x;
  const unsigned k = (unsigned)(best[t] & 0xFFFFFFFFull);
  out[IDX_OFF + t] = (float)k;
  float ss = 0.f;
  for (int d = 0; d < 128; ++d) {
    const float q = emb[k * 128 + d];
    zq[t * 128 + d] = q;
    const float diff = q - z[t * 128 + d];
    ss += diff * diff;
  }
  red[t] = ss;
  __syncthreads();
  for (int s = 128; s > 0; s >>= 1) {
    if (t < s) red[t] += red[t + s];
    __syncthreads();
  }
  if (t == 0) {
    const float L = red[0] / 32768.0f;
    out[LOSS0_OFF] = L;  // codebook_loss
    out[LOSS1_OFF] = L;  // commitment_loss (same forward value)
  }
}

// ============================================================================
// Kernel F: decoder on 8x8 canonical tile (rows/cols {0,1,2,3,60,61,62,63}),
// then broadcast to 64x64. One block per batch element. Layer 1 uses the
// 9-category pre-summed weights WS1 (constant input => one 128-dot/output).
// ============================================================================
__global__ void __launch_bounds__(256)
kdecode(const float* __restrict__ zq, const float* __restrict__ WS1,
        const float* __restrict__ db1, const float* __restrict__ wt2,
        const float* __restrict__ db2, const float* __restrict__ wt3,
        const float* __restrict__ db3, float* __restrict__ out) {
  __shared__ __align__(16) float zs[128];
  __shared__ float buf1[64 * 64];  // [pixel][chan]
  __shared__ float buf2[64 * 64];
  __shared__ float canon[192];
  const int b = blockIdx.x, t = threadIdx.x;
  if (t < 128) zs[t] = zq[b * 128 + t];
  __syncthreads();

  // layer 1: 128 -> 64 (spatially constant input; category-summed weights)
  for (int idx = t; idx < 4096; idx += 256) {
    const int o = idx >> 6, p = idx & 63;
    const int r = p >> 3, c = p & 7;
    const int cat = ((r == 0) ? 0 : ((r == 7) ? 2 : 1)) * 3 +
                    ((c == 0) ? 0 : ((c == 7) ? 2 : 1));
    const float4* wp = (const float4*)(WS1 + (cat * 64 + o) * 128);
    const float4* zp = (const float4*)zs;
    float val = db1[o];
#pragma unroll 8
    for (int i = 0; i < 32; ++i) {
      const float4 wv = wp[i], zv = zp[i];
      val = fmaf(wv.x, zv.x, val);
      val = fmaf(wv.y, zv.y, val);
      val = fmaf(wv.z, zv.z, val);
      val = fmaf(wv.w, zv.w, val);
    }
    buf1[p * 64 + o] = fmaxf(val, 0.f);
  }
  __syncthreads();

  // layer 2: 64 -> 64
  for (int idx = t; idx < 4096; idx += 256) {
    const int o = idx >> 6, p = idx & 63;
    const int r = p >> 3, c = p & 7;
    float val = db2[o];
    const int ky0 = (r == 0) ? 1 : 0, ky1 = (r == 7) ? 1 : 2;
    const int kx0 = (c == 0) ? 1 : 0, kx1 = (c == 7) ? 1 : 2;
    for (int ky = ky0; ky <= ky1; ++ky)
      for (int kx = kx0; kx <= kx1; ++kx) {
        const int pp = (r + ky - 1) * 8 + (c + kx - 1);
        const float* wv = wt2 + (o * 9 + ky * 3 + kx) * 64;
        const float* iv = buf1 + pp * 64;
#pragma unroll 16
        for (int i = 0; i < 64; ++i) val = fmaf(wv[i], iv[i], val);
      }
    buf2[p * 64 + o] = fmaxf(val, 0.f);
  }
  __syncthreads();

  // layer 3: 64 -> 3, tanh
  if (t < 192) {
    const int o = t >> 6, p = t & 63;
    const int r = p >> 3, c = p & 7;
    float val = db3[o];
    const int ky0 = (r == 0) ? 1 : 0, ky1 = (r == 7) ? 1 : 2;
    const int kx0 = (c == 0) ? 1 : 0, kx1 = (c == 7) ? 1 : 2;
    for (int ky = ky0; ky <= ky1; ++ky)
      for (int kx = kx0; kx <= kx1; ++kx) {
        const int pp = (r + ky - 1) * 8 + (c + kx - 1);
        const float* wv = wt3 + (o * 9 + ky * 3 + kx) * 64;
        const float* iv = buf2 + pp * 64;
        for (int i = 0; i < 64; ++i) val = fmaf(wv[i], iv[i], val);
      }
    canon[o * 64 + p] = tanhf(val);
  }
  __syncthreads();

  // broadcast canonical tile to full 64x64 output
  for (int q = t; q < 12288; q += 256) {
    const int o = q >> 12, p = q & 4095;
    const int r = p >> 6, c = p & 63;
    const int rm = (r < 4) ? r : ((r >= 60) ? r - 56 : 3);
    const int cm = (c < 4) ? c : ((c >= 60) ? c - 56 : 3);
    out[b * 12288 + q] = canon[o * 64 + rm * 8 + cm];
  }
}

// ============================================================================
extern "C" void kernel_launch(void* const* d_in, const int* in_sizes, int n_in,
                              void* d_out, int out_size, void* d_ws, size_t ws_size,
                              hipStream_t stream) {
  const float* patch  = (const float*)d_in[0];
  const float* enc_w1 = (const float*)d_in[1];
  const float* enc_b1 = (const float*)d_in[2];
  const float* enc_w2 = (const float*)d_in[3];
  const float* enc_b2 = (const float*)d_in[4];
  const float* emb    = (const float*)d_in[5];
  const float* dec_w1 = (const float*)d_in[6];
  const float* dec_b1 = (const float*)d_in[7];
  const float* dec_w2 = (const float*)d_in[8];
  const float* dec_b2 = (const float*)d_in[9];
  const float* dec_w3 = (const float*)d_in[10];
  const float* dec_b3 = (const float*)d_in[11];
  float* out = (float*)d_out;

  char* ws = (char*)d_ws;
  float* Sbuf  = (float*)(ws + 0);                         // 256*576
  float* W2eff = (float*)(ws + 589824);                    // 576*128
  float* z     = (float*)(ws + 884736);                    // 256*128
  float* en    = (float*)(ws + 1015808);                   // 8192
  unsigned long long* best = (unsigned long long*)(ws + 1048576);  // 256
  float* zq    = (float*)(ws + 1050624);                   // 256*128
  float* WS1   = (float*)(ws + 1181696);                   // 9*64*128
  float* wt2   = (float*)(ws + 1476608);                   // 64*9*64
  float* wt3   = (float*)(ws + 1624064);                   // 3*9*64

  kconv1_sums<<<BATCH, 256, 0, stream>>>(patch, enc_w1, enc_b1, Sbuf);
  kprep_w2<<<(576 * 128 + 255) / 256, 256, 0, stream>>>(enc_w2, W2eff);
  kprep_en<<<KCODES / 256, 256, 0, stream>>>(emb, en);
  kprep_wt<<<(73728 + 36864 + 1728 + 255) / 256, 256, 0, stream>>>(
      dec_w1, dec_w2, dec_w3, WS1, wt2, wt3);
  kinit_best<<<1, 256, 0, stream>>>(best);

  kz_gemm<<<dim3(16, 8), 32, 0, stream>>>(Sbuf, W2eff, enc_b2, z);
  kvq<<<dim3(16, 128), 32, 0, stream>>>(z, emb, en, best);
  kfinalize<<<1, 256, 0, stream>>>(best, emb, z, zq, out);
  kdecode<<<BATCH, 256, 0, stream>>>(zq, WS1, dec_b1, wt2, dec_b2, wt3, dec_b3, out);
}